// FunctionEncoder_80848464379991
// MI455X (gfx1250) — compile-verified
//
#include <hip/hip_runtime.h>
#include <math.h>

#define H 128
#define NUM_EDGE_TYPES 9
#define NUM_LAYERS 2

typedef __attribute__((ext_vector_type(2))) float v2f;
typedef __attribute__((ext_vector_type(8))) float v8f;

// ---------------------------------------------------------------------------
// Edge phase: one wave32 per edge; lane l covers features [4l, 4l+4).
// gather x[src] + edge_emb[type], scatter-add into agg[dst] with native
// global_atomic_add_f32 (L2-resident: agg is 25.6MB << 192MB L2).
// ---------------------------------------------------------------------------
__global__ __launch_bounds__(256)
void edge_scatter_kernel(const float* __restrict__ x,
                         const int* __restrict__ src,
                         const int* __restrict__ dst,
                         const int* __restrict__ etyp,
                         const float* __restrict__ eemb,
                         float* __restrict__ agg, int E) {
    const int lane = threadIdx.x & 31;
    const int gw   = (blockIdx.x * blockDim.x + threadIdx.x) >> 5;
    if (gw >= E) return;
    const int s = src[gw];
    const int d = dst[gw];
    const int t = etyp[gw];
    const float4 xv = *(const float4*)(x + (size_t)s * H + lane * 4);
    const float4 ev = *(const float4*)(eemb + (size_t)t * H + lane * 4);
    float* ap = agg + (size_t)d * H + lane * 4;
    __hip_atomic_fetch_add(ap + 0, xv.x + ev.x, __ATOMIC_RELAXED, __HIP_MEMORY_SCOPE_AGENT);
    __hip_atomic_fetch_add(ap + 1, xv.y + ev.y, __ATOMIC_RELAXED, __HIP_MEMORY_SCOPE_AGENT);
    __hip_atomic_fetch_add(ap + 2, xv.z + ev.z, __ATOMIC_RELAXED, __HIP_MEMORY_SCOPE_AGENT);
    __hip_atomic_fetch_add(ap + 3, xv.w + ev.w, __ATOMIC_RELAXED, __HIP_MEMORY_SCOPE_AGENT);
}

// ---------------------------------------------------------------------------
// Node phase: out = relu(x@sw.T + agg@mw.T + sb + mb) -> LayerNorm.
// 256 threads = 8 waves; block computes 16 rows x 128 cols. Wave w owns the
// 16x16 tile at columns [16w,16w+16) using V_WMMA_F32_16X16X4_F32, K=128 in
// steps of 4, chaining the self-path and msg-path WMMAs into one accumulator.
// A layout (16x4 f32): row = lane&15, VGPR pair = K { (lane>>4)*2, +1 }.
// B (4x16) = W.T: same addressing with row = output feature on lanes.
// D layout: VGPR v -> row v + 8*(lane>>4), col = lane&15.
// ---------------------------------------------------------------------------
__global__ __launch_bounds__(256)
void gemm_relu_ln_kernel(const float* __restrict__ xin,
                         const float* __restrict__ agg,
                         const float* __restrict__ sw,
                         const float* __restrict__ sb,
                         const float* __restrict__ mw,
                         const float* __restrict__ mb,
                         const float* __restrict__ lng,
                         const float* __restrict__ lnb,
                         float* __restrict__ xout, int N) {
    __shared__ float tile[16][H + 4];
    __shared__ float mu[16];
    __shared__ float rs[16];

    const int lane = threadIdx.x & 31;
    const int wave = threadIdx.x >> 5;
    const int row0 = blockIdx.x << 4;
    if (row0 >= N) return;

    const int m   = lane & 15;          // A row / D col / B feature row
    const int kq  = (lane >> 4) << 1;   // K sub-offset: 0 or 2
    const int col = (wave << 4) + m;    // output feature of this lane's B row

    const float* xr = xin + (size_t)(row0 + m) * H + kq;
    const float* ar = agg + (size_t)(row0 + m) * H + kq;
    const float* s1 = sw  + (size_t)col * H + kq;   // sw[col][k] = (sw.T)[k][col]
    const float* m1 = mw  + (size_t)col * H + kq;

    v8f c = {0.f, 0.f, 0.f, 0.f, 0.f, 0.f, 0.f, 0.f};
#pragma unroll
    for (int k = 0; k < H; k += 4) {
        v2f a0 = *(const v2f*)(xr + k);
        v2f b0 = *(const v2f*)(s1 + k);
        c = __builtin_amdgcn_wmma_f32_16x16x4_f32(false, a0, false, b0,
                                                  (short)0, c, false, false);
        v2f a1 = *(const v2f*)(ar + k);
        v2f b1 = *(const v2f*)(m1 + k);
        c = __builtin_amdgcn_wmma_f32_16x16x4_f32(false, a1, false, b1,
                                                  (short)0, c, false, false);
    }

    const float bias  = sb[col] + mb[col];
    const int   rbase = (lane >> 4) << 3;   // lanes 16-31 hold rows 8..15
#pragma unroll
    for (int v = 0; v < 8; ++v) {
        float val = c[v] + bias;
        tile[rbase + v][col] = val > 0.f ? val : 0.f;   // relu
    }
    __syncthreads();

    // LayerNorm stats: one thread per row over the full 128-wide row.
    if (threadIdx.x < 16) {
        float s = 0.f, q = 0.f;
#pragma unroll 8
        for (int j = 0; j < H; ++j) {
            float t = tile[threadIdx.x][j];
            s += t; q += t * t;
        }
        float mean = s * (1.0f / H);
        float var  = q * (1.0f / H) - mean * mean;
        mu[threadIdx.x] = mean;
        rs[threadIdx.x] = rsqrtf(var + 1e-5f);
    }
    __syncthreads();

    // normalize + affine + store: 2048 elements / 256 threads = 8 each
#pragma unroll
    for (int i = 0; i < 8; ++i) {
        int e = threadIdx.x * 8 + i;
        int r = e >> 7;
        int cc = e & (H - 1);
        xout[(size_t)(row0 + r) * H + cc] =
            (tile[r][cc] - mu[r]) * rs[r] * lng[cc] + lnb[cc];
    }
}

// ---------------------------------------------------------------------------
// Pooling pass 1: scores[n] = dot(x[n], attn_w) + attn_b. One wave per node.
// ---------------------------------------------------------------------------
__global__ __launch_bounds__(256)
void scores_kernel(const float* __restrict__ x,
                   const float* __restrict__ aw,
                   const float* __restrict__ ab,
                   float* __restrict__ scores, int N) {
    const int lane = threadIdx.x & 31;
    const int gw   = (blockIdx.x * blockDim.x + threadIdx.x) >> 5;
    const int nw   = (gridDim.x * blockDim.x) >> 5;
    const float4 w = *(const float4*)(aw + lane * 4);
    const float b0 = ab[0];
    for (int n = gw; n < N; n += nw) {
        const float4 xv = *(const float4*)(x + (size_t)n * H + lane * 4);
        float s = xv.x * w.x + xv.y * w.y + xv.z * w.z + xv.w * w.w;
#pragma unroll
        for (int o = 16; o > 0; o >>= 1) s += __shfl_xor(s, o, 32);
        if (lane == 0) scores[n] = s + b0;
    }
}

__global__ __launch_bounds__(256)
void score_max_partial_kernel(const float* __restrict__ scores,
                              float* __restrict__ partials, int N) {
    __shared__ float red[256];
    float m = -3.402823466e38f;
    for (int i = blockIdx.x * blockDim.x + threadIdx.x; i < N;
         i += blockDim.x * gridDim.x)
        m = fmaxf(m, scores[i]);
    red[threadIdx.x] = m;
    __syncthreads();
    for (int s = 128; s > 0; s >>= 1) {
        if (threadIdx.x < s)
            red[threadIdx.x] = fmaxf(red[threadIdx.x], red[threadIdx.x + s]);
        __syncthreads();
    }
    if (threadIdx.x == 0) partials[blockIdx.x] = red[0];
}

__global__ __launch_bounds__(256)
void score_max_final_kernel(const float* __restrict__ partials,
                            float* __restrict__ M) {
    __shared__ float red[256];
    red[threadIdx.x] = partials[threadIdx.x];
    __syncthreads();
    for (int s = 128; s > 0; s >>= 1) {
        if (threadIdx.x < s)
            red[threadIdx.x] = fmaxf(red[threadIdx.x], red[threadIdx.x + s]);
        __syncthreads();
    }
    if (threadIdx.x == 0) M[0] = red[0];
}

// ---------------------------------------------------------------------------
// Pooling pass 2: sumexp and sum_n exp(s_n - M) * x[n][:], block-staged in
// LDS (ds_add_f32) so global f32 atomics are 1 per feature per block.
// ---------------------------------------------------------------------------
__global__ __launch_bounds__(256)
void pool_accum_kernel(const float* __restrict__ x,
                       const float* __restrict__ scores,
                       const float* __restrict__ Mptr,
                       float* __restrict__ outacc,
                       float* __restrict__ sumexp, int N) {
    __shared__ float facc[H];
    __shared__ float sacc;
    if (threadIdx.x < H) facc[threadIdx.x] = 0.f;
    if (threadIdx.x == 0) sacc = 0.f;
    __syncthreads();

    const int lane = threadIdx.x & 31;
    const int wave = threadIdx.x >> 5;
    const int gw   = blockIdx.x * 8 + wave;
    const int nw   = gridDim.x * 8;
    const float M  = Mptr[0];

    float4 acc = {0.f, 0.f, 0.f, 0.f};
    float wsum = 0.f;
    for (int n = gw; n < N; n += nw) {
        const float w = expf(scores[n] - M);
        const float4 xv = *(const float4*)(x + (size_t)n * H + lane * 4);
        acc.x += w * xv.x; acc.y += w * xv.y;
        acc.z += w * xv.z; acc.w += w * xv.w;
        if (lane == 0) wsum += w;
    }
    float* fp = &facc[lane * 4];
    __hip_atomic_fetch_add(fp + 0, acc.x, __ATOMIC_RELAXED, __HIP_MEMORY_SCOPE_WORKGROUP);
    __hip_atomic_fetch_add(fp + 1, acc.y, __ATOMIC_RELAXED, __HIP_MEMORY_SCOPE_WORKGROUP);
    __hip_atomic_fetch_add(fp + 2, acc.z, __ATOMIC_RELAXED, __HIP_MEMORY_SCOPE_WORKGROUP);
    __hip_atomic_fetch_add(fp + 3, acc.w, __ATOMIC_RELAXED, __HIP_MEMORY_SCOPE_WORKGROUP);
    if (lane == 0)
        __hip_atomic_fetch_add(&sacc, wsum, __ATOMIC_RELAXED, __HIP_MEMORY_SCOPE_WORKGROUP);
    __syncthreads();

    if (threadIdx.x < H)
        __hip_atomic_fetch_add(&outacc[threadIdx.x], facc[threadIdx.x],
                               __ATOMIC_RELAXED, __HIP_MEMORY_SCOPE_AGENT);
    if (threadIdx.x == 0)
        __hip_atomic_fetch_add(sumexp, sacc, __ATOMIC_RELAXED, __HIP_MEMORY_SCOPE_AGENT);
}

__global__ void finalize_kernel(const float* __restrict__ outacc,
                                const float* __restrict__ sumexp,
                                float* __restrict__ out) {
    if (threadIdx.x < H) out[threadIdx.x] = outacc[threadIdx.x] / sumexp[0];
}

// ---------------------------------------------------------------------------
extern "C" void kernel_launch(void* const* d_in, const int* in_sizes, int n_in,
                              void* d_out, int out_size, void* d_ws, size_t ws_size,
                              hipStream_t stream) {
    const float* x0    = (const float*)d_in[0];
    const int*   eidx  = (const int*)d_in[1];
    const int*   etyp  = (const int*)d_in[2];
    const float* eemb  = (const float*)d_in[3];
    const float* selfw = (const float*)d_in[4];
    const float* selfb = (const float*)d_in[5];
    const float* msgw  = (const float*)d_in[6];
    const float* msgb  = (const float*)d_in[7];
    const float* lng   = (const float*)d_in[8];
    const float* lnb   = (const float*)d_in[9];
    const float* attnw = (const float*)d_in[10];
    const float* attnb = (const float*)d_in[11];

    const int N = in_sizes[0] / H;
    const int E = in_sizes[2];
    const int* src = eidx;
    const int* dst = eidx + E;

    float* ws       = (float*)d_ws;
    float* xb0      = ws;
    float* xb1      = ws + (size_t)N * H;
    float* agg      = ws + 2 * (size_t)N * H;
    float* scores   = ws + 3 * (size_t)N * H;
    float* partials = scores + N;
    float* Mv       = partials + 256;
    float* outacc   = Mv + 1;
    float* sumexp   = outacc + H;

    const float* xin = x0;
    float* xbufs[2] = {xb0, xb1};
    const int gemm_blocks = (N + 15) / 16;
    const int edge_blocks = (E + 7) / 8;   // one wave32 per edge, 8 waves/block

    for (int l = 0; l < NUM_LAYERS; ++l) {
        hipMemsetAsync(agg, 0, (size_t)N * H * sizeof(float), stream);
        edge_scatter_kernel<<<edge_blocks, 256, 0, stream>>>(
            xin, src, dst, etyp, eemb + (size_t)l * NUM_EDGE_TYPES * H, agg, E);
        gemm_relu_ln_kernel<<<gemm_blocks, 256, 0, stream>>>(
            xin, agg,
            selfw + (size_t)l * H * H, selfb + (size_t)l * H,
            msgw  + (size_t)l * H * H, msgb  + (size_t)l * H,
            lng + (size_t)l * H, lnb + (size_t)l * H,
            xbufs[l], N);
        xin = xbufs[l];
    }

    scores_kernel<<<256, 256, 0, stream>>>(xin, attnw, attnb, scores, N);
    score_max_partial_kernel<<<256, 256, 0, stream>>>(scores, partials, N);
    score_max_final_kernel<<<1, 256, 0, stream>>>(partials, Mv);
    hipMemsetAsync(outacc, 0, (H + 1) * sizeof(float), stream);
    pool_accum_kernel<<<256, 256, 0, stream>>>(xin, scores, Mv, outacc, sumexp, N);
    finalize_kernel<<<1, 128, 0, stream>>>(outacc, sumexp, (float*)d_out);
}